// LatentCEDisRNNAgent_56702158242170
// MI455X (gfx1250) — compile-verified
//
#include <hip/hip_runtime.h>
#include <math.h>

typedef __attribute__((ext_vector_type(2))) float v2f;
typedef __attribute__((ext_vector_type(8))) float v8f;

#define N_AG 512

// ---- workspace layout (float offsets) ----
#define OFF_T1     0u          // 512x64 encoder pre-BN
#define OFF_T2     32768u      // 512x64 inference pre-BN
#define OFF_H      65536u      // 512x64 encoder post BN+lrelu
#define OFF_HI     98304u      // 512x64 inference post BN+lrelu
#define OFF_LAT    131072u     // 512x512 lat  (mu = lat[:, :256])
#define OFF_LI     393216u     // 512x512 li
#define OFF_LATENT 655360u     // 512x256
#define OFF_I2V    786432u     // 512x256  0.5/var
#define OFF_LSIG   917504u     // 512x256  log(sig)
#define OFF_A      1048576u    // 512x64   latent @ dW1[:256]
#define OFF_B      1081344u    // 512x64   latent @ dW1[256:]
#define OFF_SUMS   1114112u    // 256: sumA,sumB,sumA2,sumB2 (64 each)
#define OFF_BNS    1114368u    // 512x64 per-i BN scale
#define OFF_BNSH   1147136u    // 512x64 per-i BN shift
#define OFF_MI     1179904u    // 512x512
#define OFF_DIS    1442048u    // 512x512
#define OFF_MM     1704192u    // 4 ordered-uint min/max
#define OFF_ENTP   1704200u    // 512 entropy partials
#define OFF_KLP    1704712u    // 512 kl partials
#define OFF_NMP    1705224u    // 512 min-term partials
#define OFF_NDP    1705736u    // 512 dis_n partials

__device__ __forceinline__ unsigned f2ord(float f) {
  unsigned u = __float_as_uint(f);
  return (u & 0x80000000u) ? ~u : (u | 0x80000000u);
}
__device__ __forceinline__ float ord2f(unsigned o) {
  unsigned u = (o & 0x80000000u) ? (o & 0x7fffffffu) : ~o;
  return __uint_as_float(u);
}

// ---------------- init ----------------
__global__ void k_init(unsigned* mm) {
  if (threadIdx.x == 0) {
    mm[0] = 0xFFFFFFFFu; mm[1] = 0u;   // mi min/max (ordered)
    mm[2] = 0xFFFFFFFFu; mm[3] = 0u;   // dis min/max (ordered)
  }
}

// ---------------- WMMA f32 GEMM: t1 = obs[:, 528:] @ eW1 + eb1 ----------------
__global__ void k_gemm_enc(const float* __restrict__ obs, const float* __restrict__ eW1,
                           const float* __restrict__ eb1, float* __restrict__ t1) {
  int tm = blockIdx.x, tn = blockIdx.y;
  int lane = threadIdx.x;
  int hf = lane >> 4;          // K sub-block 0/1
  int l  = lane & 15;
  const float* arow = obs + (tm * 16 + l) * 656 + 528;
  int col = tn * 16 + l;
  v8f acc = {};
  #pragma unroll
  for (int k = 0; k < 128; k += 4) {
    int kk = k + hf * 2;
    v2f a, b;
    a.x = arow[kk];            a.y = arow[kk + 1];
    b.x = eW1[kk * 64 + col];  b.y = eW1[(kk + 1) * 64 + col];
    acc = __builtin_amdgcn_wmma_f32_16x16x4_f32(false, a, false, b, (short)0, acc, false, false);
  }
  float bias = eb1[col];
  #pragma unroll
  for (int r = 0; r < 8; ++r) {
    int orow = tm * 16 + hf * 8 + r;
    t1[orow * 64 + col] = acc[r] + bias;
  }
}

// ---------------- WMMA f32 GEMM: t2 = [hidden|obs] @ iW1 + ib1 (K=720) ----------------
__global__ void k_gemm_inf(const float* __restrict__ hs, const float* __restrict__ obs,
                           const float* __restrict__ iW1, const float* __restrict__ ib1,
                           float* __restrict__ t2) {
  int tm = blockIdx.x, tn = blockIdx.y;
  int lane = threadIdx.x;
  int hf = lane >> 4;
  int l  = lane & 15;
  int row = tm * 16 + l;
  int col = tn * 16 + l;
  v8f acc = {};
  for (int k = 0; k < 720; k += 4) {
    int kk = k + hf * 2;
    v2f a, b;
    a.x = (kk < 64)     ? hs[row * 64 + kk]       : obs[row * 656 + (kk - 64)];
    a.y = (kk + 1 < 64) ? hs[row * 64 + kk + 1]   : obs[row * 656 + (kk - 63)];
    b.x = iW1[kk * 64 + col];
    b.y = iW1[(kk + 1) * 64 + col];
    acc = __builtin_amdgcn_wmma_f32_16x16x4_f32(false, a, false, b, (short)0, acc, false, false);
  }
  float bias = ib1[col];
  #pragma unroll
  for (int r = 0; r < 8; ++r) {
    int orow = tm * 16 + hf * 8 + r;
    t2[orow * 64 + col] = acc[r] + bias;
  }
}

// ---------------- BN over axis 0 + leaky_relu ----------------
__global__ void k_bn(const float* __restrict__ t1, const float* __restrict__ t2,
                     const float* __restrict__ eg, const float* __restrict__ ebt,
                     const float* __restrict__ ig, const float* __restrict__ ibt,
                     float* __restrict__ h, float* __restrict__ hi) {
  __shared__ float sm[256], sm2[256];
  int which = blockIdx.x >> 6;
  int c = blockIdx.x & 63;
  const float* x = which ? t2 : t1;
  float*       y = which ? hi : h;
  const float* g = which ? ig : eg;
  const float* b = which ? ibt : ebt;
  int tid = threadIdx.x;
  float v0 = x[tid * 64 + c], v1 = x[(tid + 256) * 64 + c];
  sm[tid]  = v0 + v1;
  sm2[tid] = v0 * v0 + v1 * v1;
  __syncthreads();
  for (int off = 128; off > 0; off >>= 1) {
    if (tid < off) { sm[tid] += sm[tid + off]; sm2[tid] += sm2[tid + off]; }
    __syncthreads();
  }
  float mean  = sm[0] * (1.0f / 512.0f);
  float var   = sm2[0] * (1.0f / 512.0f) - mean * mean;
  float scale = g[c] * rsqrtf(var + 1e-5f);
  float sh    = b[c] - mean * scale;
  float y0 = fmaf(v0, scale, sh); y0 = y0 > 0.f ? y0 : 0.01f * y0;
  float y1 = fmaf(v1, scale, sh); y1 = y1 > 0.f ? y1 : 0.01f * y1;
  y[tid * 64 + c] = y0;
  y[(tid + 256) * 64 + c] = y1;
}

// ---------------- WMMA f32 GEMM: lat = h@eW2+eb2 / li = hi@iW2+ib2 (K=64,N=512) ----------------
__global__ void k_gemm_latli(const float* __restrict__ h, const float* __restrict__ hi,
                             const float* __restrict__ eW2, const float* __restrict__ eb2,
                             const float* __restrict__ iW2, const float* __restrict__ ib2,
                             float* __restrict__ lat, float* __restrict__ li) {
  const float *Ain, *W, *bias; float* outp;
  if (blockIdx.z == 0) { Ain = h;  W = eW2; bias = eb2; outp = lat; }
  else                 { Ain = hi; W = iW2; bias = ib2; outp = li;  }
  int tm = blockIdx.x, tn = blockIdx.y;
  int lane = threadIdx.x;
  int hf = lane >> 4;
  int l  = lane & 15;
  const float* arow = Ain + (tm * 16 + l) * 64;
  int col = tn * 16 + l;
  v8f acc = {};
  #pragma unroll
  for (int k = 0; k < 64; k += 4) {
    int kk = k + hf * 2;
    v2f a, b;
    a.x = arow[kk];             a.y = arow[kk + 1];
    b.x = W[kk * 512 + col];    b.y = W[(kk + 1) * 512 + col];
    acc = __builtin_amdgcn_wmma_f32_16x16x4_f32(false, a, false, b, (short)0, acc, false, false);
  }
  float bv = bias[col];
  #pragma unroll
  for (int r = 0; r < 8; ++r) {
    int orow = tm * 16 + hf * 8 + r;
    outp[orow * 512 + col] = acc[r] + bv;
  }
}

// ---------------- pointwise: var/sig/latent + entropy & KL partials ----------------
__global__ void k_post(const float* __restrict__ lat, const float* __restrict__ li,
                       const float* __restrict__ eps, float* __restrict__ latent,
                       float* __restrict__ i2v, float* __restrict__ lsig,
                       float* __restrict__ entp, float* __restrict__ klp) {
  __shared__ float se[256], sk[256];
  int tid = threadIdx.x;
  int idx = blockIdx.x * 256 + tid;       // 0..131071
  int m = idx >> 8, c = idx & 255;
  float mu  = lat[m * 512 + c];
  float lv  = lat[m * 512 + 256 + c];
  float var = fmaxf(expf(lv), 0.002f);
  float logsig = 0.5f * logf(var);
  float sig = sqrtf(var);
  latent[idx] = fmaf(sig, eps[idx], mu);
  i2v[idx]    = 0.5f / var;
  lsig[idx]   = logsig;
  float muq = li[m * 512 + c];
  float lvq = li[m * 512 + 256 + c];
  float varq = fmaxf(expf(lvq), 0.002f);
  float logsigq = 0.5f * logf(varq);
  float d = mu - muq;
  se[tid] = 1.41893853f + logsig;  // 0.5 + 0.5*log(2pi) + log(sig)
  sk[tid] = (logsigq - logsig) + (var + d * d) * (0.5f / varq) - 0.5f;
  __syncthreads();
  for (int off = 128; off > 0; off >>= 1) {
    if (tid < off) { se[tid] += se[tid + off]; sk[tid] += sk[tid + off]; }
    __syncthreads();
  }
  if (tid == 0) { entp[blockIdx.x] = se[0]; klp[blockIdx.x] = sk[0]; }
}

// ---------------- WMMA f32 GEMM: A/B = latent @ dW1 halves (K=256,N=64) ----------------
__global__ void k_gemm_ab(const float* __restrict__ latent, const float* __restrict__ dW1,
                          float* __restrict__ A, float* __restrict__ B) {
  const float* W = dW1 + blockIdx.z * 256 * 64;
  float* outp    = blockIdx.z ? B : A;
  int tm = blockIdx.x, tn = blockIdx.y;
  int lane = threadIdx.x;
  int hf = lane >> 4;
  int l  = lane & 15;
  const float* arow = latent + (tm * 16 + l) * 256;
  int col = tn * 16 + l;
  v8f acc = {};
  #pragma unroll 8
  for (int k = 0; k < 256; k += 4) {
    int kk = k + hf * 2;
    v2f a, b;
    a.x = arow[kk];           a.y = arow[kk + 1];
    b.x = W[kk * 64 + col];   b.y = W[(kk + 1) * 64 + col];
    acc = __builtin_amdgcn_wmma_f32_16x16x4_f32(false, a, false, b, (short)0, acc, false, false);
  }
  #pragma unroll
  for (int r = 0; r < 8; ++r) {
    int orow = tm * 16 + hf * 8 + r;
    outp[orow * 64 + col] = acc[r];
  }
}

// ---------------- per-channel moments of A and B ----------------
__global__ void k_colsum(const float* __restrict__ A, const float* __restrict__ B,
                         float* __restrict__ sums) {
  __shared__ float sa[256], sb[256], sa2[256], sb2[256];
  int c = blockIdx.x, tid = threadIdx.x;
  float a0 = A[tid * 64 + c], a1 = A[(tid + 256) * 64 + c];
  float b0 = B[tid * 64 + c], b1 = B[(tid + 256) * 64 + c];
  sa[tid]  = a0 + a1;          sb[tid]  = b0 + b1;
  sa2[tid] = a0 * a0 + a1 * a1; sb2[tid] = b0 * b0 + b1 * b1;
  __syncthreads();
  for (int off = 128; off > 0; off >>= 1) {
    if (tid < off) {
      sa[tid] += sa[tid + off];  sb[tid] += sb[tid + off];
      sa2[tid] += sa2[tid + off]; sb2[tid] += sb2[tid + off];
    }
    __syncthreads();
  }
  if (tid == 0) { sums[c] = sa[0]; sums[64 + c] = sb[0]; sums[128 + c] = sa2[0]; sums[192 + c] = sb2[0]; }
}

// ---------------- circular cross-corr -> per-i BN scale/shift (axis-1 BN) ----------------
__global__ void k_corr(const float* __restrict__ A, const float* __restrict__ B,
                       const float* __restrict__ sums, const float* __restrict__ dg,
                       const float* __restrict__ dbt, float* __restrict__ bns,
                       float* __restrict__ bnsh) {
  __shared__ float sm[256];
  int i = blockIdx.x, tid = threadIdx.x;
  int c = tid & 63, s = tid >> 6;
  float acc = 0.f;
  #pragma unroll 4
  for (int jj = 0; jj < 128; ++jj) {
    int j = s * 128 + jj;
    int k = (j - i + 511) & 511;
    acc = fmaf(A[j * 64 + c], B[k * 64 + c], acc);
  }
  sm[tid] = acc;
  __syncthreads();
  if (tid < 64) {
    float C = sm[tid] + sm[tid + 64] + sm[tid + 128] + sm[tid + 192];
    float mean = (sums[tid] + sums[64 + tid]) * (1.0f / 512.0f);
    float e2   = (sums[128 + tid] + sums[192 + tid] + 2.0f * C) * (1.0f / 512.0f);
    float var  = e2 - mean * mean;
    float scale = dg[tid] * rsqrtf(var + 1e-5f);
    bns[i * 64 + tid]  = scale;
    bnsh[i * 64 + tid] = dbt[tid] - mean * scale;   // db1 cancels in BN
  }
}

// ---------------- fused mi + dis pass with global min/max tracking ----------------
__global__ void k_midis(const float* __restrict__ latentp, const float* __restrict__ lat,
                        const float* __restrict__ i2v, const float* __restrict__ lsig,
                        const float* __restrict__ A, const float* __restrict__ B,
                        const float* __restrict__ bns, const float* __restrict__ bnsh,
                        const float* __restrict__ dW2, const float* __restrict__ db2,
                        float* __restrict__ mi, float* __restrict__ dis, unsigned* __restrict__ mm) {
  __shared__ float sbs[64], sbh[64], sw2[64];
  int i = blockIdx.x, tid = threadIdx.x;
  if (tid < 64) { sbs[tid] = bns[i * 64 + tid]; sbh[tid] = bnsh[i * 64 + tid]; sw2[tid] = dW2[tid]; }
  __syncthreads();
  int wv = tid >> 5, lane = tid & 31;
  float db2v = db2[0];
  float lmin_mi = 3.4e38f, lmax_mi = -3.4e38f, lmin_d = 3.4e38f, lmax_d = -3.4e38f;
  for (int j = wv; j < 512; j += 8) {
    int k = (j - i + 511) & 511;
    const float* lk = latentp + k * 256;
    const float* mj = lat + j * 512;
    const float* vj = i2v + j * 256;
    const float* sj = lsig + j * 256;
    float msum = 0.f;
    #pragma unroll
    for (int cc = 0; cc < 8; ++cc) {
      int c = lane + cc * 32;
      float d = lk[c] - mj[c];
      float t = fmaf(-d * d, vj[c], -sj[c] - 0.91893853f);
      msum += fmaxf(t + 13.9f, -13.9f);
    }
    float dsum = 0.f;
    #pragma unroll
    for (int cc = 0; cc < 2; ++cc) {
      int c = lane + cc * 32;
      float hb = fmaf(sbs[c], A[j * 64 + c] + B[k * 64 + c], sbh[c]);
      hb = hb > 0.f ? hb : 0.01f * hb;
      dsum = fmaf(hb, sw2[c], dsum);
    }
    #pragma unroll
    for (int off = 16; off > 0; off >>= 1) {
      msum += __shfl_xor(msum, off, 32);
      dsum += __shfl_xor(dsum, off, 32);
    }
    float mi_v  = msum * (1.0f / 256.0f);
    float dis_v = fabsf(dsum + db2v);
    if (lane == 0) { mi[i * 512 + j] = mi_v; dis[i * 512 + j] = dis_v; }
    lmin_mi = fminf(lmin_mi, mi_v); lmax_mi = fmaxf(lmax_mi, mi_v);
    lmin_d  = fminf(lmin_d, dis_v); lmax_d  = fmaxf(lmax_d, dis_v);
  }
  if (lane == 0) {
    atomicMin(&mm[0], f2ord(lmin_mi));
    atomicMax(&mm[1], f2ord(lmax_mi));
    atomicMin(&mm[2], f2ord(lmin_d));
    atomicMax(&mm[3], f2ord(lmax_d));
  }
}

// ---------------- normalized sums (deterministic two-stage) ----------------
__global__ void k_norm(const float* __restrict__ mi, const float* __restrict__ dis,
                       const unsigned* __restrict__ mm, float* __restrict__ nmp,
                       float* __restrict__ ndp) {
  __shared__ float s1[256], s2[256];
  float mi_min = ord2f(mm[0]), mi_max = ord2f(mm[1]);
  float d_min  = ord2f(mm[2]), d_max  = ord2f(mm[3]);
  float inv_mi = 1.0f / (mi_max - mi_min + 1e-12f);
  float inv_d  = 1.0f / (d_max - d_min + 1e-12f);
  int tid = threadIdx.x;
  float a1 = 0.f, a2 = 0.f;
  for (int idx = blockIdx.x * 256 + tid; idx < 262144; idx += 512 * 256) {
    float mn = (mi[idx] - mi_min) * inv_mi;
    float dn = (dis[idx] - d_min) * inv_d;
    a1 += fminf(mn + dn, 1.0f);
    a2 += dn;
  }
  s1[tid] = a1; s2[tid] = a2;
  __syncthreads();
  for (int off = 128; off > 0; off >>= 1) {
    if (tid < off) { s1[tid] += s1[tid + off]; s2[tid] += s2[tid + off]; }
    __syncthreads();
  }
  if (tid == 0) { nmp[blockIdx.x] = s1[0]; ndp[blockIdx.x] = s2[0]; }
}

// ---------------- final scalars ----------------
__global__ void k_final(const float* __restrict__ entp, const float* __restrict__ klp,
                        const float* __restrict__ nmp, const float* __restrict__ ndp,
                        float* __restrict__ out) {
  if (threadIdx.x != 0) return;
  float es = 0.f, ks = 0.f, ns = 0.f, ds = 0.f;
  for (int b = 0; b < 512; ++b) { es += entp[b]; ks += klp[b]; ns += nmp[b]; ds += ndp[b]; }
  float entropy = es * (1.0f / 512.0f);
  float kl      = ks * (1.0f / 512.0f);
  float loss0 = fminf(entropy * 1e-4f + kl * 1e-4f, 2000.0f);
  float ce = log1pf(expf(loss0));
  float dis_loss = -ns * (1.0f / 512.0f);
  float dis_norm =  ds * (1.0f / 512.0f);
  float c_dis = (dis_norm + dis_loss) * (1.0f / 512.0f);
  out[0] = ce + c_dis;
  out[1] = c_dis;
  out[2] = ce;
}

extern "C" void kernel_launch(void* const* d_in, const int* in_sizes, int n_in,
                              void* d_out, int out_size, void* d_ws, size_t ws_size,
                              hipStream_t stream) {
  (void)in_sizes; (void)n_in; (void)out_size; (void)ws_size;
  const float* obs    = (const float*)d_in[0];
  const float* hidden = (const float*)d_in[1];
  const float* eps    = (const float*)d_in[2];
  const float* eW1    = (const float*)d_in[3];
  const float* eb1    = (const float*)d_in[4];
  const float* eg     = (const float*)d_in[5];
  const float* ebt    = (const float*)d_in[6];
  const float* eW2    = (const float*)d_in[7];
  const float* eb2    = (const float*)d_in[8];
  const float* iW1    = (const float*)d_in[9];
  const float* ib1    = (const float*)d_in[10];
  const float* ig     = (const float*)d_in[11];
  const float* ibt    = (const float*)d_in[12];
  const float* iW2    = (const float*)d_in[13];
  const float* ib2    = (const float*)d_in[14];
  const float* dW1    = (const float*)d_in[15];
  // d_in[16] = db1 : cancels exactly under axis-1 BatchNorm
  const float* dg     = (const float*)d_in[17];
  const float* dbt    = (const float*)d_in[18];
  const float* dW2    = (const float*)d_in[19];
  const float* db2    = (const float*)d_in[20];

  float* ws = (float*)d_ws;
  float* out = (float*)d_out;
  unsigned* mm = (unsigned*)(ws + OFF_MM);

  k_init<<<1, 32, 0, stream>>>(mm);
  k_gemm_enc<<<dim3(32, 4), 32, 0, stream>>>(obs, eW1, eb1, ws + OFF_T1);
  k_gemm_inf<<<dim3(32, 4), 32, 0, stream>>>(hidden, obs, iW1, ib1, ws + OFF_T2);
  k_bn<<<128, 256, 0, stream>>>(ws + OFF_T1, ws + OFF_T2, eg, ebt, ig, ibt,
                                ws + OFF_H, ws + OFF_HI);
  k_gemm_latli<<<dim3(32, 32, 2), 32, 0, stream>>>(ws + OFF_H, ws + OFF_HI, eW2, eb2, iW2, ib2,
                                                   ws + OFF_LAT, ws + OFF_LI);
  k_post<<<512, 256, 0, stream>>>(ws + OFF_LAT, ws + OFF_LI, eps, ws + OFF_LATENT,
                                  ws + OFF_I2V, ws + OFF_LSIG, ws + OFF_ENTP, ws + OFF_KLP);
  k_gemm_ab<<<dim3(32, 4, 2), 32, 0, stream>>>(ws + OFF_LATENT, dW1, ws + OFF_A, ws + OFF_B);
  k_colsum<<<64, 256, 0, stream>>>(ws + OFF_A, ws + OFF_B, ws + OFF_SUMS);
  k_corr<<<512, 256, 0, stream>>>(ws + OFF_A, ws + OFF_B, ws + OFF_SUMS, dg, dbt,
                                  ws + OFF_BNS, ws + OFF_BNSH);
  k_midis<<<512, 256, 0, stream>>>(ws + OFF_LATENT, ws + OFF_LAT, ws + OFF_I2V, ws + OFF_LSIG,
                                   ws + OFF_A, ws + OFF_B, ws + OFF_BNS, ws + OFF_BNSH,
                                   dW2, db2, ws + OFF_MI, ws + OFF_DIS, mm);
  k_norm<<<512, 256, 0, stream>>>(ws + OFF_MI, ws + OFF_DIS, mm, ws + OFF_NMP, ws + OFF_NDP);
  k_final<<<1, 32, 0, stream>>>(ws + OFF_ENTP, ws + OFF_KLP, ws + OFF_NMP, ws + OFF_NDP, out);
}